// Integrator_20048907338336
// MI455X (gfx1250) — compile-verified
//
#include <hip/hip_runtime.h>
#include <cstdint>

#define N_SAMP 192
#define CHUNKS 6            // 192 / 32 samples per lane
#define WAVES_PER_BLOCK 8
#define RAYS_PER_WAVE 4

// Issue one ray's raw tile (192 x float4 = 3072 B) as 6 async b128 DMA ops,
// one 512B fully-coalesced transfer per chunk. src_base already includes lane.
__device__ __forceinline__ void issue_async_ray(const float4* src_base, uint32_t lds_base) {
#pragma unroll
    for (int c = 0; c < CHUNKS; ++c) {
        unsigned long long g = (unsigned long long)(uintptr_t)(src_base + c * 32);
        uint32_t l = lds_base + (uint32_t)(c * 32 * 16);
        asm volatile("global_load_async_to_lds_b128 %0, %1, off"
                     :: "v"(l), "v"(g)
                     : "memory");
    }
}

__global__ __launch_bounds__(256) void nerf_integrate(
    const float* __restrict__ raw, const float* __restrict__ zvals,
    const float* __restrict__ rays_d, float* __restrict__ out,
    int n_rays, int wstride)
{
    __shared__ float4 sraw[WAVES_PER_BLOCK][2][N_SAMP];   // 48 KB: 2 buffers / wave

    const int lane = threadIdx.x & 31;
    const int wid  = threadIdx.x >> 5;
    const int wgid = blockIdx.x * WAVES_PER_BLOCK + wid;

    const float4* raw4 = (const float4*)raw;

    int ray = wgid;
    if (ray < n_rays) {
        issue_async_ray(raw4 + (size_t)ray * N_SAMP + lane,
                        (uint32_t)(uintptr_t)(void*)&sraw[wid][0][lane]);
    }

    int it = 0;
    for (; ray < n_rays; ray += wstride, ++it) {
        const int buf = it & 1;
        const int nxt = ray + wstride;
        const bool has_next = (nxt < n_rays);

        if (has_next) {
            // prefetch next ray into the other buffer, then wait for the
            // current 6 transfers (async ops complete in order -> cnt<=6).
            issue_async_ray(raw4 + (size_t)nxt * N_SAMP + lane,
                            (uint32_t)(uintptr_t)(void*)&sraw[wid][buf ^ 1][lane]);
            asm volatile("s_wait_asynccnt 0x6" ::: "memory");
        } else {
            asm volatile("s_wait_asynccnt 0x0" ::: "memory");
        }

        // ray direction norm (wave-uniform)
        float dx = rays_d[3 * ray + 0];
        float dy = rays_d[3 * ray + 1];
        float dz = rays_d[3 * ray + 2];
        float dnorm = sqrtf(dx * dx + dy * dy + dz * dz);

        const float* zb = zvals + (size_t)ray * N_SAMP;

        float carry  = 1.0f;   // product of q over all previous chunks
        float acc_w  = 0.f, acc_wz = 0.f;
        float acc_c0 = 0.f, acc_c1 = 0.f, acc_c2 = 0.f;
        float acc_wlw = 0.f;   // sum w*log(w)

#pragma unroll
        for (int c = 0; c < CHUNKS; ++c) {
            const int s = c * 32 + lane;
            float z = zb[s];
            int s1 = (s < N_SAMP - 1) ? s + 1 : s;          // clamp: no OOB
            float zn = zb[s1];
            float dist = ((s == N_SAMP - 1) ? 1e10f : (zn - z)) * dnorm;

            float4 rv = sraw[wid][buf][s];                  // ds_load_b128
            float sg0 = 1.f / (1.f + __expf(-rv.x));
            float sg1 = 1.f / (1.f + __expf(-rv.y));
            float sg2 = 1.f / (1.f + __expf(-rv.z));
            float a   = fmaxf(rv.w, 0.f) * dist;
            float e   = __expf(-a);
            float alpha = 1.f - e;
            float q   = e + 1e-10f;                         // 1 - alpha + 1e-10

            // inclusive product-scan of q across the wave (log2(32) steps)
            float p = q;
#pragma unroll
            for (int d = 1; d < 32; d <<= 1) {
                float t = __shfl_up(p, (unsigned)d, 32);
                if (lane >= d) p *= t;
            }
            // exclusive within chunk + cross-chunk carry = transmittance
            float excl = __shfl_up(p, 1u, 32);
            if (lane == 0) excl = 1.0f;
            float T = carry * excl;
            float w = alpha * T;

            acc_w   += w;
            acc_wz  += w * z;
            acc_c0  += w * sg0;
            acc_c1  += w * sg1;
            acc_c2  += w * sg2;
            acc_wlw += (w > 0.f) ? (w * __logf(w)) : 0.f;

            carry *= __shfl(p, 31, 32);                     // chunk total product
        }

        // wave-wide reductions (butterfly)
#pragma unroll
        for (int d = 16; d > 0; d >>= 1) {
            acc_w   += __shfl_xor(acc_w,   d, 32);
            acc_wz  += __shfl_xor(acc_wz,  d, 32);
            acc_c0  += __shfl_xor(acc_c0,  d, 32);
            acc_c1  += __shfl_xor(acc_c1,  d, 32);
            acc_c2  += __shfl_xor(acc_c2,  d, 32);
            acc_wlw += __shfl_xor(acc_wlw, d, 32);
        }

        if (lane == 0) {
            // sum(probs) = acc_w + (1 - acc_w + 1e-6) = 1 + 1e-6 exactly
            const float Z    = 1.0f + 1e-6f;
            const float logZ = __logf(Z);
            float sum_plogp  = (acc_wlw - acc_w * logZ) / Z;
            float L  = 1.0f - acc_w + 1e-6f;                // leftover mass
            float tL = (L > 0.f) ? (L / Z) * (__logf(L) - logZ) : 0.f;
            float sparsity = -(sum_plogp + tL);
            float depth = acc_wz / acc_w;

            out[3 * (size_t)ray + 0] = acc_c0;              // chs_map [N,3]
            out[3 * (size_t)ray + 1] = acc_c1;
            out[3 * (size_t)ray + 2] = acc_c2;
            out[(size_t)n_rays * 3 + ray] = depth;          // depth_map [N]
            out[(size_t)n_rays * 4 + ray] = sparsity;       // sparsity  [N]
        }
    }
}

extern "C" void kernel_launch(void* const* d_in, const int* in_sizes, int n_in,
                              void* d_out, int out_size, void* d_ws, size_t ws_size,
                              hipStream_t stream) {
    const float* raw    = (const float*)d_in[0];   // [N, 192, 4]
    const float* zvals  = (const float*)d_in[1];   // [N, 192]
    const float* rays_d = (const float*)d_in[2];   // [N, 3]
    float* out = (float*)d_out;                    // [N*3 | N | N]

    int n_rays = in_sizes[1] / N_SAMP;
    int waves  = (n_rays + RAYS_PER_WAVE - 1) / RAYS_PER_WAVE;
    int blocks = (waves + WAVES_PER_BLOCK - 1) / WAVES_PER_BLOCK;
    if (blocks < 1) blocks = 1;
    int wstride = blocks * WAVES_PER_BLOCK;

    nerf_integrate<<<blocks, 256, 0, stream>>>(raw, zvals, rays_d, out, n_rays, wstride);
}